// ContactMapDistError_47519518163580
// MI455X (gfx1250) — compile-verified
//
#include <hip/hip_runtime.h>

#define B  4
#define N  10475
#define R  48
#define VR 96

typedef __attribute__((ext_vector_type(2))) float v2f;
typedef __attribute__((ext_vector_type(4))) float v4f;
typedef __attribute__((ext_vector_type(8))) float v8f;

// workspace layout (float offsets)
#define OFF_A1  0                          // {-2x,-2y,-2z, 1.0}  gathered g1, [B][R][VR][4]
#define OFF_A2  (B*R*VR*4)                 // { x,  y,  z, |g2|^2} gathered g2, [B][R][VR][4]
#define OFF_SQ1 (2*B*R*VR*4)               // |g1|^2               [B][R][VR]
#define OFF_ACC (2*B*R*VR*4 + B*R*VR)      // sums[B], counts[B]

// Stage 1: gather + pack epilogue terms into the WMMA operands.
__global__ __launch_bounds__(256)
void gather_kernel(const float* __restrict__ v1s, const float* __restrict__ v2s,
                   const int* __restrict__ rid_to_vid, float* __restrict__ ws) {
    int tid = blockIdx.x * blockDim.x + threadIdx.x;
    if (tid < 2 * B) ws[OFF_ACC + tid] = 0.0f;          // zero sums+counts each call
    if (tid >= B * R * VR) return;
    int b  = tid / (R * VR);
    int rv = tid - b * (R * VR);                         // r*VR + v
    int vid = rid_to_vid[rv];

    const float* p1 = v1s + (size_t)(b * N + vid) * 3;
    float x = p1[0], y = p1[1], z = p1[2];
    // A row: K=0..2 = -2*g1 (exact scale), K=3 = 1.0 (picks up sq2 row of B)
    v4f a1 = { -2.0f * x, -2.0f * y, -2.0f * z, 1.0f };
    *(v4f*)(ws + OFF_A1 + (size_t)tid * 4) = a1;
    ws[OFF_SQ1 + tid] = x * x + y * y + z * z;

    const float* p2 = v2s + (size_t)(b * N + vid) * 3;
    x = p2[0]; y = p2[1]; z = p2[2];
    // B col: K=0..2 = g2, K=3 = |g2|^2  =>  A x B = -2*inner + sq2
    v4f a2 = { x, y, z, x * x + y * y + z * z };
    *(v4f*)(ws + OFF_A2 + (size_t)tid * 4) = a2;
}

// Stage 2: one wave32 per (b, r, s).
// D = (-2*g1 | 1) x (g2 | sq2) + C(sq1)  ==  d2 tile directly; epilogue = 8 mins.
__global__ __launch_bounds__(32)
void pairmin_kernel(const unsigned char* __restrict__ cmaps,
                    const float* __restrict__ ws,
                    float* __restrict__ acc) {
    int bid = blockIdx.x;                 // b*R*R + r*R + s
    int b  = bid / (R * R);
    int rs = bid - b * (R * R);
    int r  = rs / R;
    int s  = rs - r * R;

    // scalar (wave-uniform) cmap test -> EXEC stays all-ones for WMMA
    int flag = __builtin_amdgcn_readfirstlane((int)cmaps[bid]);
    if (!flag) return;

    int lane = threadIdx.x;
    int lo = lane & 15;                   // row/col index within tile
    int hi = lane >> 4;                   // K-half select for A/B, M-half for C/D

    const float* A1 = ws + OFF_A1 + (size_t)((b * R + r) * VR) * 4;
    const float* A2 = ws + OFF_A2 + (size_t)((b * R + s) * VR) * 4;
    const float* S1 = ws + OFF_SQ1 + (b * R + r) * VR;

    // Preload all 6 B fragments (w-tiles).
    v2f bfrag[6];
#pragma unroll
    for (int t = 0; t < 6; ++t)
        bfrag[t] = *(const v2f*)(A2 + (16 * t + lo) * 4 + 2 * hi);

    v8f minacc;
#pragma unroll
    for (int j = 0; j < 8; ++j) minacc[j] = 3.4e38f;

#pragma unroll
    for (int tv = 0; tv < 6; ++tv) {
        v2f afrag = *(const v2f*)(A1 + (16 * tv + lo) * 4 + 2 * hi);
        // C fragment = sq1 for this lane's 8 M-rows: contiguous 32B block (2x b128)
        v8f cfrag = *(const v8f*)(S1 + 16 * tv + 8 * hi);
#pragma unroll
        for (int tw = 0; tw < 6; ++tw) {
            v8f d = __builtin_amdgcn_wmma_f32_16x16x4_f32(
                false, afrag, false, bfrag[tw], (short)0, cfrag, false, false);
#pragma unroll
            for (int j = 0; j < 8; ++j)
                minacc[j] = fminf(minacc[j], d[j]);       // D is d2 already
        }
    }

    float m = minacc[0];
#pragma unroll
    for (int j = 1; j < 8; ++j) m = fminf(m, minacc[j]);
#pragma unroll
    for (int off = 16; off >= 1; off >>= 1)
        m = fminf(m, __shfl_xor(m, off, 32));

    if (lane == 0) {
        // sqrt after min (monotone): one sqrt per active (r,s)
        atomicAdd(&acc[b],     sqrtf(fmaxf(m, 0.0f)));
        atomicAdd(&acc[B + b], 1.0f);
    }
}

// Stage 3: masked mean per batch.
__global__ __launch_bounds__(32)
void finalize_kernel(const float* __restrict__ acc, float* __restrict__ out) {
    int t = threadIdx.x;
    if (t < B) out[t] = acc[t] / acc[B + t];
}

extern "C" void kernel_launch(void* const* d_in, const int* in_sizes, int n_in,
                              void* d_out, int out_size, void* d_ws, size_t ws_size,
                              hipStream_t stream) {
    const float*         v1s        = (const float*)d_in[0];
    const float*         v2s        = (const float*)d_in[1];
    const unsigned char* cmaps      = (const unsigned char*)d_in[2];  // jnp bool = 1 byte
    const int*           rid_to_vid = (const int*)d_in[3];
    float* ws  = (float*)d_ws;
    float* acc = ws + OFF_ACC;

    int total = B * R * VR;                                    // 18432 gather threads
    gather_kernel<<<(total + 255) / 256, 256, 0, stream>>>(v1s, v2s, rid_to_vid, ws);
    pairmin_kernel<<<B * R * R, 32, 0, stream>>>(cmaps, ws, acc);  // 9216 wave32 blocks
    finalize_kernel<<<1, 32, 0, stream>>>(acc, (float*)d_out);
}